// Embedding2Y_42073499632158
// MI455X (gfx1250) — compile-verified
//
#include <hip/hip_runtime.h>

// Anti-diagonal means of a [B=64, T=512, E=8192] f32 tensor, keeping
// diagonals s = T .. T+E-2  ->  out [64, 8191].
//
// Memory-bound: ~1.07 GB streamed exactly once (roofline ~46 us @ 23.3 TB/s).
// Row reduction runs on the matrix pipe via V_WMMA_F32_16X16X4_F32 with
// A = ones (exact f32 accumulation, RNE). One wave owns 16 consecutive
// diagonals; at row r those 16 elements are contiguous in memory
// (flat offset s + r*(E-1)), so every load is coalesced and every fetched
// byte is payload.
//
// v2 changes vs v1 (asm showed exec-branch predication + loadcnt-0 before
// every WMMA):
//  - loads are unconditional (addresses provably stay inside the batch slab;
//    the single phantom-lane overflow is fixed by clamping s), bounds are
//    applied as branchless v_cndmask on the loaded value,
//  - wave-uniform dense/tail split: 480/512 groups need no masking at all,
//  - 32 rows per iteration: 16 loads issued back-to-back (immediate offsets),
//    then 8 chained WMMAs -> 16 loads in flight per wave.

typedef __attribute__((ext_vector_type(2))) float v2f;
typedef __attribute__((ext_vector_type(8))) float v8f;

constexpr int kB = 64;
constexpr int kT = 512;
constexpr int kE = 8192;
constexpr int kOut = kE - 1;               // 8191 outputs per batch
constexpr int kFullCntMaxJ = kE - kT - 1;  // 7679: last j with count == T

template <bool DENSE>
__device__ __forceinline__ v8f accumulate(const float* __restrict__ xb,
                                          int offA, int offB, int half,
                                          int rminEff) {
  v8f acc = {};
  const v2f ones = {1.0f, 1.0f};
  const int step = 4 * (kE - 1);  // elements between successive 4-row groups

#pragma unroll 1
  for (int r = 0; r < kT; r += 32) {
    v2f bv[8];
#pragma unroll
    for (int u = 0; u < 8; ++u) {          // 16 loads, no predication
      bv[u].x = xb[offA + u * step];
      bv[u].y = xb[offB + u * step];
    }
    if (!DENSE) {
#pragma unroll
      for (int u = 0; u < 8; ++u) {        // branchless column-bound mask
        const int r0 = r + u * 4 + half * 2;
        bv[u].x = (r0 >= rminEff) ? bv[u].x : 0.0f;
        bv[u].y = (r0 + 1 >= rminEff) ? bv[u].y : 0.0f;
      }
    }
#pragma unroll
    for (int u = 0; u < 8; ++u) {          // D = ones(16x4)*B(4x16) + C
      acc = __builtin_amdgcn_wmma_f32_16x16x4_f32(
          /*neg_a=*/false, ones, /*neg_b=*/false, bv[u],
          /*c_mod=*/(short)0, acc, /*reuse_a=*/false, /*reuse_b=*/false);
    }
    offA += 8 * step;
    offB += 8 * step;
  }
  return acc;
}

__global__ __launch_bounds__(256) void antidiag_means_wmma(
    const float* __restrict__ x, float* __restrict__ out) {
  const int lane  = threadIdx.x & 31;
  const int wid   = threadIdx.x >> 5;
  const int gwave = blockIdx.x * (blockDim.x >> 5) + wid;  // 0 .. 32767

  const int b    = gwave >> 9;      // 512 wave-groups per batch
  const int grp  = gwave & 511;
  const int j0   = grp << 4;        // 16 diagonals per wave
  const int n    = lane & 15;       // WMMA column owned by this lane
  const int half = lane >> 4;       // which row-pair this lane feeds
  const int j    = j0 + n;          // output diagonal index
  // Clamp the phantom lane (j==8191) so its r=511 address stays in-bounds;
  // its contribution is masked to zero in the tail path anyway.
  const int jj   = (j < kOut) ? j : (kOut - 1);
  const int s    = kT + jj;         // r + c == s

  const float* xb = x + (size_t)b * kT * kE;

  // Diagonal element of row r lives at flat offset s + r*(E-1).
  // half==0 supplies rows {r, r+1}; half==1 supplies rows {r+2, r+3}.
  const int offA = s + (half * 2) * (kE - 1);
  const int offB = offA + (kE - 1);

  v8f acc;
  if (j0 + 15 <= kFullCntMaxJ) {
    // Dense: every row's column is in range for all 16 diagonals (480/512).
    acc = accumulate<true>(xb, offA, offB, half, /*rminEff=*/0);
  } else {
    // Tail: rows below rmin read past column E-1 -> mask to zero.
    const int rminEff = (j < kOut) ? (j - kFullCntMaxJ) : 0x7FFFFFFF;
    acc = accumulate<false>(xb, offA, offB, half, rminEff);
  }

  // D layout: VGPR0, lanes 0-15 hold (M=0, N=lane); all M rows identical.
  if (lane < 16 && j < kOut) {
    const float cnt = (j <= kFullCntMaxJ) ? (float)kT : (float)(kOut - j);
    out[(size_t)b * kOut + j] = acc[0] / cnt;
  }
}

extern "C" void kernel_launch(void* const* d_in, const int* in_sizes, int n_in,
                              void* d_out, int out_size, void* d_ws, size_t ws_size,
                              hipStream_t stream) {
  const float* x = (const float*)d_in[0];
  float* out = (float*)d_out;

  // 64 batches * 512 wave-groups = 32768 waves; 8 waves (256 thr) per block.
  const int total_waves = kB * (kE / 16);
  dim3 block(256);
  dim3 grid(total_waves / 8);  // 4096
  antidiag_means_wmma<<<grid, block, 0, stream>>>(x, out);
}